// MyLoc1d_71219147702608
// MI455X (gfx1250) — compile-verified
//
#include <hip/hip_runtime.h>
#include <stdint.h>

// Locally-connected 1D layer for MI455X (gfx1250), wave32 + WMMA f32 16x16x4.
// out[b,o,p] = (1/32) * sum_{c,f} x[b,c,p+f] * filt[o,c,p,f]
// One workgroup per patch position p => 32x128x1024 GEMM per p.
// CDNA5 paths: v_wmma_f32_16x16x4_f32 + global_load_async_to_lds_{b32,b128}
// with double-buffered LDS and s_wait_asynccnt pipelining.

typedef __attribute__((ext_vector_type(2))) float v2f;
typedef __attribute__((ext_vector_type(8))) float v8f;

#define B_     32
#define C_     128
#define O_     128
#define P_     1017
#define F_     8
#define KC     32      // k-chunk: 4 channels x 8 taps
#define NCHUNK 32      // 1024 / 32
#define LDS_S  36      // padded LDS row stride (dwords): conflict-free frag reads

__device__ __forceinline__ void async_to_lds_b128(uint32_t lds_off, const float* gaddr) {
    asm volatile("global_load_async_to_lds_b128 %0, %1, off"
                 :: "v"(lds_off), "v"(gaddr) : "memory");
}
__device__ __forceinline__ void async_to_lds_b32(uint32_t lds_off, const float* gaddr) {
    asm volatile("global_load_async_to_lds_b32 %0, %1, off"
                 :: "v"(lds_off), "v"(gaddr) : "memory");
}

__global__ __launch_bounds__(256)
void loc1d_wmma_async(const float* __restrict__ X,
                      const float* __restrict__ W,
                      float* __restrict__ Out) {
    __shared__ float Alds[2][B_ * LDS_S];   // [buf][b][k]  2 x 4.5 KB
    __shared__ float Blds[2][O_ * LDS_S];   // [buf][o][k]  2 x 18 KB

    const int p    = blockIdx.x;
    const int t    = threadIdx.x;
    const int lane = t & 31;
    const int wave = t >> 5;
    const int o0   = wave * 16;             // this wave's 16-wide o tile
    const int n    = lane & 15;
    const int dk   = (lane >= 16) ? 2 : 0;

    v8f acc0 = {};  // b = 0..15
    v8f acc1 = {};  // b = 16..31

    // Stage one k-chunk (async, memory -> LDS, 8 ops/thread: 4x b32 + 4x b128)
    auto stage = [&](int cc, int buf) {
        const int c0 = cc * 4;
        // A: x[b, c0+cl, p+f] -> Alds[buf][b][k], k = cl*8+f  (1024 floats)
        #pragma unroll
        for (int it = 0; it < 4; ++it) {
            int e = it * 256 + t;              // 0..1023
            int b = e >> 5, k = e & 31;
            int c = c0 + (k >> 3), f = k & 7;
            uint32_t la = (uint32_t)(size_t)&Alds[buf][b * LDS_S + k];
            async_to_lds_b32(la, X + ((size_t)(b * C_ + c) << 10) + p + f);
        }
        // B: filt[o, c0+cl, p, 0..7] -> Blds[buf][o][cl*8+f]  (1024 float4,
        // 32B-aligned global rows, 16B-aligned LDS dest)
        #pragma unroll
        for (int it = 0; it < 4; ++it) {
            int e = it * 256 + t;              // 0..1023
            int h = e & 1, R = e >> 1;         // R: 0..511
            int o = R >> 2, cl = R & 3;
            uint32_t lb = (uint32_t)(size_t)&Blds[buf][o * LDS_S + cl * 8 + h * 4];
            async_to_lds_b128(lb, W + ((size_t)(o * C_ + c0 + cl) * P_ + p) * F_ + h * 4);
        }
    };

    stage(0, 0);
    for (int cc = 0; cc < NCHUNK; ++cc) {
        const int buf = cc & 1;
        if (cc + 1 < NCHUNK) {
            stage(cc + 1, buf ^ 1);
            // 8 newer ops in flight; async loads complete in order, so
            // waiting <=8 guarantees this chunk's 8 ops have landed in LDS.
            asm volatile("s_wait_asynccnt 0x8" ::: "memory");
        } else {
            asm volatile("s_wait_asynccnt 0x0" ::: "memory");
        }
        __syncthreads();

        const float* Ab = Alds[buf];
        const float* Bb = Blds[buf];
        #pragma unroll
        for (int kt = 0; kt < 8; ++kt) {
            const int k0 = kt * 4 + dk;
            // B frag: lane L: VGPR0=B[k0][o0+n], VGPR1=B[k0+1][o0+n] (k0+2 for L>=16)
            v2f bf = *(const v2f*)&Bb[(o0 + n) * LDS_S + k0];
            // A frags: lane L: row m = L&15, two consecutive k
            v2f a0 = *(const v2f*)&Ab[n * LDS_S + k0];
            v2f a1 = *(const v2f*)&Ab[(16 + n) * LDS_S + k0];
            acc0 = __builtin_amdgcn_wmma_f32_16x16x4_f32(
                false, a0, false, bf, (short)0, acc0, false, false);
            acc1 = __builtin_amdgcn_wmma_f32_16x16x4_f32(
                false, a1, false, bf, (short)0, acc1, false, false);
        }
        __syncthreads();
    }

    // Epilogue: scale by (C*F)^-1/2 = 1/32 exactly, scatter to out[b][o][p].
    const float s  = 0.03125f;
    const int   mo = (lane >= 16) ? 8 : 0;
    const int   oc = o0 + n;
    #pragma unroll
    for (int r = 0; r < 8; ++r) {
        int b0 = r + mo;
        Out[(size_t)(b0 * O_ + oc) * P_ + p]        = acc0[r] * s;
        Out[(size_t)((b0 + 16) * O_ + oc) * P_ + p] = acc1[r] * s;
    }
}

extern "C" void kernel_launch(void* const* d_in, const int* in_sizes, int n_in,
                              void* d_out, int out_size, void* d_ws, size_t ws_size,
                              hipStream_t stream) {
    const float* x    = (const float*)d_in[0];
    const float* filt = (const float*)d_in[1];
    float*       out  = (float*)d_out;
    loc1d_wmma_async<<<dim3(P_), dim3(256), 0, stream>>>(x, filt, out);
}